// Model_paths_44349832298741
// MI455X (gfx1250) — compile-verified
//
#include <hip/hip_runtime.h>
#include <hip/hip_bf16.h>
#include <math.h>

// Problem constants (match reference)
#define NN 50000
#define IN_DIM 256
#define HH 256
#define TT 4
#define AA 6
#define EE 500000
#define MM 2
#define ATTN_D 64
#define CC 8
#define LN_EPS 1e-5f

typedef float v2f __attribute__((ext_vector_type(2)));
typedef float v8f __attribute__((ext_vector_type(8)));

// ---------------------------------------------------------------------------
// WMMA helpers: V_WMMA_F32_16X16X4_F32 (fp32 matrix core path on gfx1250)
// A: 16x4 fp32 -> 2 VGPRs/lane.  Lanes 0-15 hold K={k,k+1}, lanes 16-31 K={k+2,k+3}, M = lane&15.
// B: 4x16 fp32 -> 2 VGPRs/lane.  N = lane&15, same K split as A.
// C/D: 16x16 fp32 -> 8 VGPRs/lane. VGPR i: M = i + (lane>=16 ? 8:0), N = lane&15.
// ---------------------------------------------------------------------------
__device__ __forceinline__ v8f wmma_f32(v2f a, v2f b, v8f c) {
  return __builtin_amdgcn_wmma_f32_16x16x4_f32(false, a, false, b, (short)0, c,
                                               false, false);
}

__device__ __forceinline__ v2f load_a4(const float* __restrict__ A, int m, int k,
                                       int lane) {
  const int kb = (lane >> 4) << 1;
  const float* p = A + (size_t)m * 256 + (k + kb);
  v2f r;
  r.x = p[0];
  r.y = p[1];
  return r;
}

__device__ __forceinline__ v2f load_b4(const float* __restrict__ B, int k, int n,
                                       int lane) {
  const int kb = (lane >> 4) << 1;
  v2f r;
  r.x = B[(size_t)(k + kb) * HH + n];
  r.y = B[(size_t)(k + kb + 1) * HH + n];
  return r;
}

__device__ __forceinline__ float wave_sum(float v) {
  v += __shfl_xor(v, 16, 32);
  v += __shfl_xor(v, 8, 32);
  v += __shfl_xor(v, 4, 32);
  v += __shfl_xor(v, 2, 32);
  v += __shfl_xor(v, 1, 32);
  return v;
}

// ---------------------------------------------------------------------------
// 1) Per-node-type input projection: hid[n] = feats[n] @ wsW[type[n]] + wsb[type[n]]
// Wave computes a 16x16 tile for ALL 4 types (4 accumulators), selects at store.
// Grid: (ceil(N/16), H/(8*16)=2), block 256 (8 waves, consecutive col tiles).
// ---------------------------------------------------------------------------
__global__ __launch_bounds__(256) void proj_select_kernel(
    const float* __restrict__ feats, const int* __restrict__ types,
    const float* __restrict__ wsW, const float* __restrict__ wsb,
    float* __restrict__ hid) {
  const int lane = threadIdx.x & 31;
  const int wave = threadIdx.x >> 5;
  const int row0 = blockIdx.x * 16;
  const int col0 = (blockIdx.y * 8 + wave) * 16;
  const int m = row0 + (lane & 15);
  const int mc = m < NN ? m : (NN - 1);  // clamp: keeps EXEC full, store guarded
  const int n = col0 + (lane & 15);

  v8f c0 = {}, c1 = {}, c2 = {}, c3 = {};
#pragma unroll 4
  for (int k = 0; k < IN_DIM; k += 4) {
    v2f af = load_a4(feats, mc, k, lane);
    v2f b0 = load_b4(wsW + 0 * (IN_DIM * HH), k, n, lane);
    v2f b1 = load_b4(wsW + 1 * (IN_DIM * HH), k, n, lane);
    v2f b2 = load_b4(wsW + 2 * (IN_DIM * HH), k, n, lane);
    v2f b3 = load_b4(wsW + 3 * (IN_DIM * HH), k, n, lane);
    c0 = wmma_f32(af, b0, c0);
    c1 = wmma_f32(af, b1, c1);
    c2 = wmma_f32(af, b2, c2);
    c3 = wmma_f32(af, b3, c3);
  }

  const int half = (lane >> 4) << 3;  // 0 or 8
#pragma unroll
  for (int i = 0; i < 8; ++i) {
    const int mm = row0 + half + i;
    if (mm < NN) {
      const int t = types[mm];
      float v = (t == 0) ? c0[i] : (t == 1) ? c1[i] : (t == 2) ? c2[i] : c3[i];
      hid[(size_t)mm * HH + n] = v + wsb[t * HH + n];
    }
  }
}

// ---------------------------------------------------------------------------
// 2) Affine GEMM: Y = X @ W + bias, X:[N,256] W:[256,256].
// Wave tile 16x64 (A-frag reused across 4 column subtiles).
// Grid: (ceil(N/128), 256/64=4), block 256 (8 row-waves).
// ---------------------------------------------------------------------------
__global__ __launch_bounds__(256) void gemm_aff_kernel(
    const float* __restrict__ X, const float* __restrict__ W,
    const float* __restrict__ bias, float* __restrict__ Y) {
  const int lane = threadIdx.x & 31;
  const int wave = threadIdx.x >> 5;
  const int row0 = blockIdx.x * 128 + wave * 16;
  const int col0 = blockIdx.y * 64;
  const int m = row0 + (lane & 15);
  const int mc = m < NN ? m : (NN - 1);
  const int nl = lane & 15;

  v8f c0 = {}, c1 = {}, c2 = {}, c3 = {};
#pragma unroll 4
  for (int k = 0; k < HH; k += 4) {
    v2f af = load_a4(X, mc, k, lane);
    c0 = wmma_f32(af, load_b4(W, k, col0 + nl, lane), c0);
    c1 = wmma_f32(af, load_b4(W, k, col0 + 16 + nl, lane), c1);
    c2 = wmma_f32(af, load_b4(W, k, col0 + 32 + nl, lane), c2);
    c3 = wmma_f32(af, load_b4(W, k, col0 + 48 + nl, lane), c3);
  }

  const int half = (lane >> 4) << 3;
#pragma unroll
  for (int i = 0; i < 8; ++i) {
    const int mm = row0 + half + i;
    if (mm < NN) {
      float* yr = Y + (size_t)mm * HH;
      yr[col0 + nl] = c0[i] + bias[col0 + nl];
      yr[col0 + 16 + nl] = c1[i] + bias[col0 + 16 + nl];
      yr[col0 + 32 + nl] = c2[i] + bias[col0 + 32 + nl];
      yr[col0 + 48 + nl] = c3[i] + bias[col0 + 48 + nl];
    }
  }
}

// ---------------------------------------------------------------------------
// 3) SpMM scatter: y[r] += scale * v * x[c] over 2 selected adjacency matrices.
// Adjacency selectors read on-device (idxp[0], idxp[1]).
// One wave per edge slot; lane handles 8 columns (2x float4 gather + 8 f32 atomics).
// x (51.2MB) fits in the 192MB L2 -> gathers/atomics stay on-chip.
// ---------------------------------------------------------------------------
__global__ __launch_bounds__(256) void spmm_kernel(
    const int* __restrict__ rows, const int* __restrict__ cols,
    const float* __restrict__ vals, const int* __restrict__ idxp,
    const float* __restrict__ x, float* __restrict__ y, float scale) {
  const int w = (int)((blockIdx.x * 256u + threadIdx.x) >> 5);
  const int lane = threadIdx.x & 31;
  if (w >= 2 * EE) return;
  const int k = (w >= EE) ? 1 : 0;
  const int e = w - k * EE;
  const int adj = idxp[k];
  const size_t base = (size_t)adj * EE + e;
  const int r = rows[base];
  const int c = cols[base];
  const float v = vals[base] * scale;

  const float4* xp = (const float4*)(x + (size_t)c * HH);
  const float4 x0 = xp[lane * 2 + 0];
  const float4 x1 = xp[lane * 2 + 1];
  float* yp = y + (size_t)r * HH + lane * 8;
  unsafeAtomicAdd(yp + 0, v * x0.x);
  unsafeAtomicAdd(yp + 1, v * x0.y);
  unsafeAtomicAdd(yp + 2, v * x0.z);
  unsafeAtomicAdd(yp + 3, v * x0.w);
  unsafeAtomicAdd(yp + 4, v * x1.x);
  unsafeAtomicAdd(yp + 5, v * x1.y);
  unsafeAtomicAdd(yp + 6, v * x1.z);
  unsafeAtomicAdd(yp + 7, v * x1.w);
}

// ---------------------------------------------------------------------------
// 4) LayerNorm + exact GELU, in place. One wave per row.
// ---------------------------------------------------------------------------
__global__ __launch_bounds__(256) void ln_gelu_kernel(float* __restrict__ x,
                                                      const float* __restrict__ g,
                                                      const float* __restrict__ b) {
  const int row = (int)((blockIdx.x * 256u + threadIdx.x) >> 5);
  const int lane = threadIdx.x & 31;
  if (row >= NN) return;
  float* rp = x + (size_t)row * HH;
  float4 v0 = ((const float4*)rp)[lane * 2 + 0];
  float4 v1 = ((const float4*)rp)[lane * 2 + 1];
  float vv[8] = {v0.x, v0.y, v0.z, v0.w, v1.x, v1.y, v1.z, v1.w};

  float s = 0.f, ss = 0.f;
#pragma unroll
  for (int j = 0; j < 8; ++j) {
    s += vv[j];
    ss += vv[j] * vv[j];
  }
  s = wave_sum(s);
  ss = wave_sum(ss);
  const float mu = s * (1.0f / HH);
  const float var = ss * (1.0f / HH) - mu * mu;
  const float inv = rsqrtf(var + LN_EPS);

#pragma unroll
  for (int j = 0; j < 8; ++j) {
    const int col = lane * 8 + j;
    const float yn = (vv[j] - mu) * inv * g[col] + b[col];
    vv[j] = 0.5f * yn * (1.0f + erff(yn * 0.70710678118654752f));
  }
  float4 o0 = {vv[0], vv[1], vv[2], vv[3]};
  float4 o1 = {vv[4], vv[5], vv[6], vv[7]};
  ((float4*)rp)[lane * 2 + 0] = o0;
  ((float4*)rp)[lane * 2 + 1] = o1;
}

// ---------------------------------------------------------------------------
// 5) Attention score: scores[n,m] = tanh(hm[n] @ W1 + b1) @ W2 + b2.
// One wave per node; lane owns attention dims {lane, lane+32}.
// ---------------------------------------------------------------------------
__global__ __launch_bounds__(256) void attn_kernel(
    const float* __restrict__ hm, const float* __restrict__ W1,
    const float* __restrict__ b1, const float* __restrict__ W2,
    const float* __restrict__ b2, float* __restrict__ scores, int mcol) {
  const int node = (int)((blockIdx.x * 256u + threadIdx.x) >> 5);
  const int lane = threadIdx.x & 31;
  if (node >= NN) return;
  const float* xr = hm + (size_t)node * HH;
  const int d0 = lane, d1 = lane + 32;
  float a0 = 0.f, a1 = 0.f;
  for (int h = 0; h < HH; ++h) {
    const float xv = xr[h];
    a0 += xv * W1[h * ATTN_D + d0];
    a1 += xv * W1[h * ATTN_D + d1];
  }
  a0 = tanhf(a0 + b1[d0]);
  a1 = tanhf(a1 + b1[d1]);
  float p = a0 * W2[d0] + a1 * W2[d1];
  p = wave_sum(p);
  if (lane == 0) scores[node * MM + mcol] = p + b2[0];
}

// ---------------------------------------------------------------------------
// 6) 2-way softmax blend + classifier: out[n] = (a0*hm0[n] + a1*hm1[n]) @ clsW + clsb
// One wave per node; register-space reduction, no LDS.
// ---------------------------------------------------------------------------
__global__ __launch_bounds__(256) void combine_cls_kernel(
    const float* __restrict__ hm0, const float* __restrict__ hm1,
    const float* __restrict__ scores, const float* __restrict__ clsW,
    const float* __restrict__ clsb, float* __restrict__ out) {
  const int node = (int)((blockIdx.x * 256u + threadIdx.x) >> 5);
  const int lane = threadIdx.x & 31;
  if (node >= NN) return;
  const float s0 = scores[node * MM + 0];
  const float s1 = scores[node * MM + 1];
  const float mx = fmaxf(s0, s1);
  const float e0 = __expf(s0 - mx);
  const float e1 = __expf(s1 - mx);
  const float a0 = e0 / (e0 + e1);
  const float a1 = 1.0f - a0;

  const int h0 = lane * 8;
  const float4* p0 = (const float4*)(hm0 + (size_t)node * HH + h0);
  const float4* p1 = (const float4*)(hm1 + (size_t)node * HH + h0);
  float4 u0 = p0[0], u1 = p0[1], w0 = p1[0], w1 = p1[1];
  float comb[8] = {a0 * u0.x + a1 * w0.x, a0 * u0.y + a1 * w0.y,
                   a0 * u0.z + a1 * w0.z, a0 * u0.w + a1 * w0.w,
                   a0 * u1.x + a1 * w1.x, a0 * u1.y + a1 * w1.y,
                   a0 * u1.z + a1 * w1.z, a0 * u1.w + a1 * w1.w};

  float acc[CC];
#pragma unroll
  for (int c = 0; c < CC; ++c) acc[c] = 0.f;
#pragma unroll
  for (int j = 0; j < 8; ++j) {
    const float* wr = clsW + (size_t)(h0 + j) * CC;
#pragma unroll
    for (int c = 0; c < CC; ++c) acc[c] += comb[j] * wr[c];
  }
#pragma unroll
  for (int c = 0; c < CC; ++c) acc[c] = wave_sum(acc[c]);
  if (lane == 0) {
#pragma unroll
    for (int c = 0; c < CC; ++c) out[(size_t)node * CC + c] = acc[c] + clsb[c];
  }
}

// ---------------------------------------------------------------------------
// Host orchestration
// ---------------------------------------------------------------------------
extern "C" void kernel_launch(void* const* d_in, const int* in_sizes, int n_in,
                              void* d_out, int out_size, void* d_ws, size_t ws_size,
                              hipStream_t stream) {
  const float* feats = (const float*)d_in[0];
  const int* types = (const int*)d_in[1];
  const int* arows = (const int*)d_in[2];
  const int* acols = (const int*)d_in[3];
  const float* avals = (const float*)d_in[4];
  const int* idx_seq = (const int*)d_in[5];  // [M,S,K] = [2,2,2]
  const int* idx_res = (const int*)d_in[6];  // [M,1,K] = [2,1,2]
  const float* wsW = (const float*)d_in[7];
  const float* wsb = (const float*)d_in[8];
  const float* affW = (const float*)d_in[9];
  const float* affb = (const float*)d_in[10];
  const float* lng = (const float*)d_in[11];
  const float* lnb = (const float*)d_in[12];
  const float* a1W = (const float*)d_in[13];
  const float* a1b = (const float*)d_in[14];
  const float* a2W = (const float*)d_in[15];
  const float* a2b = (const float*)d_in[16];
  const float* clsW = (const float*)d_in[17];
  const float* clsb = (const float*)d_in[18];
  float* out = (float*)d_out;

  char* ws = (char*)d_ws;
  const size_t NHb = (size_t)NN * HH * sizeof(float);
  float* hid = (float*)(ws + 0 * NHb);
  float* s0 = (float*)(ws + 1 * NHb);
  float* s1 = (float*)(ws + 2 * NHb);
  float* hm0 = (float*)(ws + 3 * NHb);
  float* hm1 = (float*)(ws + 4 * NHb);
  float* scores = (float*)(ws + 5 * NHb);

  const int per_node_waves = (NN * 32 + 255) / 256;  // 6250 blocks
  const int spmm_blocks = (2 * EE * 32 + 255) / 256; // 125000 blocks

  // Per-type projection + type-select (WMMA)
  proj_select_kernel<<<dim3((NN + 15) / 16, HH / 128), 256, 0, stream>>>(
      feats, types, wsW, wsb, hid);

  for (int m = 0; m < MM; ++m) {
    float* hm = m ? hm1 : hm0;
    // s0 = hid @ affW[m] + affb[m]  (WMMA)
    gemm_aff_kernel<<<dim3((NN + 127) / 128, HH / 64), 256, 0, stream>>>(
        hid, affW + (size_t)m * HH * HH, affb + m * HH, s0);
    // s1 = spmm(idx_seq[m][0], s0) / 2
    hipMemsetAsync(s1, 0, NHb, stream);
    spmm_kernel<<<spmm_blocks, 256, 0, stream>>>(arows, acols, avals,
                                                 idx_seq + (m * 2 + 0) * 2, s0, s1,
                                                 0.5f);
    // hm = spmm(idx_seq[m][1], s1)/2 + RATIO * spmm(idx_res[m][0], s0)/2
    hipMemsetAsync(hm, 0, NHb, stream);
    spmm_kernel<<<spmm_blocks, 256, 0, stream>>>(arows, acols, avals,
                                                 idx_seq + (m * 2 + 1) * 2, s1, hm,
                                                 0.5f);
    spmm_kernel<<<spmm_blocks, 256, 0, stream>>>(arows, acols, avals,
                                                 idx_res + m * 2, s0, hm, 0.5f);
    // hm = gelu(layernorm(hm))  in place
    ln_gelu_kernel<<<per_node_waves, 256, 0, stream>>>(hm, lng + m * HH,
                                                       lnb + m * HH);
    // scores[:, m]
    attn_kernel<<<per_node_waves, 256, 0, stream>>>(hm, a1W, a1b, a2W, a2b, scores,
                                                    m);
  }

  combine_cls_kernel<<<per_node_waves, 256, 0, stream>>>(hm0, hm1, scores, clsW,
                                                         clsb, out);
}